// DAWN_7361573945461
// MI455X (gfx1250) — compile-verified
//
#include <hip/hip_runtime.h>
#include <math.h>

// ---------------------------------------------------------------------------
// Model constants (from the reference)
// ---------------------------------------------------------------------------
#define Vv     32000
#define Dd     512
#define DFFd   2048
#define Ll     4
#define Hh     8
#define Nn     64
#define Kk     8
#define NBb    32
#define Rr     64
#define Bb     2
#define Ss     512
#define DHd    64
#define TOK    (Bb * Ss)     // 1024 tokens

typedef __attribute__((ext_vector_type(2))) float v2f;
typedef __attribute__((ext_vector_type(8))) float v8f;

// ---------------------------------------------------------------------------
// Generic fp32 WMMA GEMM:  C = epilogue( scale * A@B + bias + addend )
//   - A is MxK row-major (lda), B is KxN row-major (ldb) or, if transB,
//     stored as NxK row-major (ldb) and used as B[k][n] = Bt[n][k].
//   - Batched via blockIdx.z with two-level strides: base += (z/HH)*s1+(z%HH)*s2
//   - mode 0: none; 1: exact GELU; 2: causal mask (col>row -> -1e30, bias/addend
//     ignored).  scale_ptr (device) overrides scale if non-null.
// Requires M%64==0, N%64==0, K%32==0, all ld's multiples of 4, 16B-aligned bases.
// ---------------------------------------------------------------------------
struct GemmP {
  const float* A; long lda; long sA1; long sA2;
  const float* B; long ldb; long sB1; long sB2;
  const float* bias;
  const float* addend;           // same layout as C (ldc / batch strides)
  float* C; long ldc; long sC1; long sC2;
  int M, N, Kd, HH, transB, mode;
  float scale;
  const float* scale_ptr;
};

__global__ __launch_bounds__(256) void gemm_f32_wmma(GemmP p) {
  constexpr int BM = 64, BN = 64, BK = 32, LDT = 36;  // 36 floats = 144B rows
  __shared__ float As[BM * LDT];   // [m][k]
  __shared__ float Bs[BN * LDT];   // [n][k]  (transposed stage)
  const int  z  = blockIdx.z;
  const long zb = z / p.HH, zh = z % p.HH;
  const float* A  = p.A + zb * p.sA1 + zh * p.sA2;
  const float* Bg = p.B + zb * p.sB1 + zh * p.sB2;
  float*       C  = p.C + zb * p.sC1 + zh * p.sC2;
  const float* addend = p.addend ? (p.addend + zb * p.sC1 + zh * p.sC2) : nullptr;

  const int m0   = blockIdx.x * BM;
  const int n0   = blockIdx.y * BN;
  const int tid  = threadIdx.x;
  const int lane = tid & 31;
  const int wave = tid >> 5;           // 8 waves
  const int half = lane >> 4;          // hi/lo half-wave
  const int l    = lane & 15;
  const int m_off = (wave & 3) * 16;   // 4 M sub-tiles
  const int n_off = (wave >> 2) * 32;  // 2 N groups of 32

  v8f c0 = {0.f, 0.f, 0.f, 0.f, 0.f, 0.f, 0.f, 0.f};
  v8f c1 = c0;

  for (int k0 = 0; k0 < p.Kd; k0 += BK) {
    // ---- stage A tile 64x32 (512 float4 loads, 2 per thread) ----
#pragma unroll
    for (int it = 0; it < 2; ++it) {
      int f = tid + it * 256;
      int r = f >> 3, c4 = (f & 7) << 2;
      const float4 av = *(const float4*)(A + (long)(m0 + r) * p.lda + (k0 + c4));
      *(float4*)&As[r * LDT + c4] = av;
    }
    // ---- stage B tile as Bs[n][k] ----
    if (p.transB) {
#pragma unroll
      for (int it = 0; it < 2; ++it) {
        int f = tid + it * 256;
        int r = f >> 3, c4 = (f & 7) << 2;
        const float4 bv = *(const float4*)(Bg + (long)(n0 + r) * p.ldb + (k0 + c4));
        *(float4*)&Bs[r * LDT + c4] = bv;
      }
    } else {
#pragma unroll
      for (int it = 0; it < 2; ++it) {
        int f = tid + it * 256;
        int kk = f >> 4, c4 = (f & 15) << 2;
        const float4 bv = *(const float4*)(Bg + (long)(k0 + kk) * p.ldb + (n0 + c4));
        Bs[(c4 + 0) * LDT + kk] = bv.x;
        Bs[(c4 + 1) * LDT + kk] = bv.y;
        Bs[(c4 + 2) * LDT + kk] = bv.z;
        Bs[(c4 + 3) * LDT + kk] = bv.w;
      }
    }
    __syncthreads();
    // ---- 8 K-steps of 4, two v_wmma_f32_16x16x4_f32 per step ----
#pragma unroll
    for (int ks = 0; ks < BK; ks += 4) {
      const int kb = ks + 2 * half;  // A/B VGPR0:{k,k+2}, VGPR1:{k+1,k+3}
      v2f a  = *(const v2f*)&As[(m_off + l) * LDT + kb];
      v2f b0 = *(const v2f*)&Bs[(n_off + l) * LDT + kb];
      v2f b1 = *(const v2f*)&Bs[(n_off + 16 + l) * LDT + kb];
      c0 = __builtin_amdgcn_wmma_f32_16x16x4_f32(false, a, false, b0, (short)0, c0, false, false);
      c1 = __builtin_amdgcn_wmma_f32_16x16x4_f32(false, a, false, b1, (short)0, c1, false, false);
    }
    __syncthreads();
  }

  const float scale = p.scale_ptr ? p.scale_ptr[0] : p.scale;
#pragma unroll
  for (int i = 0; i < 8; ++i) {
    const int gr = m0 + m_off + i + 8 * half;   // C VGPR i -> rows i / i+8
#pragma unroll
    for (int jt = 0; jt < 2; ++jt) {
      const int gc = n0 + n_off + jt * 16 + l;
      float t = (jt ? c1[i] : c0[i]) * scale;
      if (p.mode == 2) {
        if (gc > gr) t = -1.0e30f;              // causal mask
      } else {
        if (p.bias)   t += p.bias[gc];
        if (addend)   t += addend[(long)gr * p.ldc + gc];
        if (p.mode == 1) t = 0.5f * t * (1.0f + erff(t * 0.70710678118654752440f));
      }
      C[(long)gr * p.ldc + gc] = t;
    }
  }
}

// ---------------------------------------------------------------------------
// Elementwise / reduction kernels
// ---------------------------------------------------------------------------
__global__ void embed_kernel(const int* ids, const float* tok, const float* pos, float* x) {
  int idx = blockIdx.x * 256 + threadIdx.x;           // < TOK*Dd
  int t = idx >> 9, d = idx & 511;
  int s = t & (Ss - 1);
  x[idx] = tok[(long)ids[t] * Dd + d] + pos[(long)s * Dd + d];
}

__global__ __launch_bounds__(256) void ln_kernel(const float* x, const float* g,
                                                 const float* b, float* out) {
  __shared__ float red[256];
  const int   tid = threadIdx.x;
  const float* xr = x + (long)blockIdx.x * Dd;
  float v0 = xr[tid], v1 = xr[tid + 256];
  red[tid] = v0 + v1;
  __syncthreads();
  for (int off = 128; off > 0; off >>= 1) {
    if (tid < off) red[tid] += red[tid + off];
    __syncthreads();
  }
  const float mean = red[0] * (1.0f / 512.0f);
  __syncthreads();
  float d0 = v0 - mean, d1 = v1 - mean;
  red[tid] = d0 * d0 + d1 * d1;
  __syncthreads();
  for (int off = 128; off > 0; off >>= 1) {
    if (tid < off) red[tid] += red[tid + off];
    __syncthreads();
  }
  const float inv = rsqrtf(red[0] * (1.0f / 512.0f) + 1e-5f);
  float* o = out + (long)blockIdx.x * Dd;
  o[tid]       = d0 * inv * g[tid]       + b[tid];
  o[tid + 256] = d1 * inv * g[tid + 256] + b[tid + 256];
}

__global__ __launch_bounds__(256) void softmax512_kernel(float* a) {
  __shared__ float red[256];
  const int tid = threadIdx.x;
  float* r = a + (long)blockIdx.x * Ss;
  float v0 = r[tid], v1 = r[tid + 256];
  red[tid] = fmaxf(v0, v1);
  __syncthreads();
  for (int off = 128; off > 0; off >>= 1) {
    if (tid < off) red[tid] = fmaxf(red[tid], red[tid + off]);
    __syncthreads();
  }
  const float mx = red[0];
  __syncthreads();
  float e0 = expf(v0 - mx), e1 = expf(v1 - mx);
  red[tid] = e0 + e1;
  __syncthreads();
  for (int off = 128; off > 0; off >>= 1) {
    if (tid < off) red[tid] += red[tid + off];
    __syncthreads();
  }
  const float inv = 1.0f / red[0];
  r[tid] = e0 * inv;
  r[tid + 256] = e1 * inv;
}

__global__ void softrec_kernel(const float* rec, float* softrec) {  // <<<64,32>>>
  int n = blockIdx.x, t = threadIdx.x;
  float v = rec[n * NBb + t];
  float m = v;
  for (int off = 16; off; off >>= 1) m = fmaxf(m, __shfl_xor(m, off));
  float e = expf(v - m);
  float s = e;
  for (int off = 16; off; off >>= 1) s += __shfl_xor(s, off);
  softrec[n * NBb + t] = e / s;
}

__global__ void ne_kernel(const float* softrec, const float* basis_emb, float* ne) {
  int idx = blockIdx.x * 256 + threadIdx.x;  // < Nn*Dd
  int n = idx >> 9, d = idx & 511;
  float acc = 0.f;
  for (int nb = 0; nb < NBb; ++nb) acc += softrec[n * NBb + nb] * basis_emb[nb * Dd + d];
  ne[idx] = acc;
}

__global__ void topk_kernel(const float* scores, float* weights, int* topidx) {
  int t = blockIdx.x * 256 + threadIdx.x;    // < TOK
  const float* s = scores + (long)t * Nn;
  float v[Nn];
  for (int j = 0; j < Nn; ++j) v[j] = s[j];
  float tv[Kk]; int ti[Kk];
  for (int kk = 0; kk < Kk; ++kk) {
    float best = -3.4e38f; int bi = 0;
    for (int j = 0; j < Nn; ++j)
      if (v[j] > best) { best = v[j]; bi = j; }   // ties -> lowest index
    tv[kk] = best; ti[kk] = bi; v[bi] = -3.4e38f;
  }
  const float mx = tv[0];
  float e[Kk], sum = 0.f;
  for (int kk = 0; kk < Kk; ++kk) { e[kk] = expf(tv[kk] - mx); sum += e[kk]; }
  const float inv = 1.0f / sum;
  for (int kk = 0; kk < Kk; ++kk) {
    weights[(long)t * Kk + kk] = e[kk] * inv;
    topidx[(long)t * Kk + kk]  = ti[kk];
  }
}

__global__ void token_recipe_kernel(const float* weights, const int* topidx,
                                    const float* softrec, float* tr) {
  int idx = blockIdx.x * 256 + threadIdx.x;  // < TOK*NBb
  int t = idx >> 5, nb = idx & 31;
  float acc = 0.f;
  for (int kk = 0; kk < Kk; ++kk)
    acc += weights[(long)t * Kk + kk] * softrec[topidx[(long)t * Kk + kk] * NBb + nb];
  tr[idx] = acc;
}

__global__ void h_kernel(const float* tr, const float* proj, float* h) {
  int idx = blockIdx.x * 256 + threadIdx.x;  // < TOK*Rr
  int t = idx >> 6, r = idx & 63;
  float acc = 0.f;
  for (int n = 0; n < NBb; ++n)
    acc += tr[t * NBb + n] * proj[(long)t * (NBb * Rr) + n * Rr + r];
  h[idx] = acc;
}

__global__ void g_kernel(const float* tr, const float* h, float* g) {
  long idx = (long)blockIdx.x * 256 + threadIdx.x;  // < TOK*NBb*Rr
  int t = (int)(idx >> 11);
  int nr = (int)(idx & 2047);
  int n = nr >> 6, r = nr & 63;
  g[idx] = tr[t * NBb + n] * h[t * Rr + r];
}

__global__ void basisA_transpose_kernel(const float* basis_A, float* A1, float* A2) {
  int idx = blockIdx.x * 256 + threadIdx.x;  // < NBb*Dd*Rr
  int n = idx >> 15;
  int rem = idx & 32767;
  int d = rem >> 6, r = rem & 63;
  float v = basis_A[idx];                    // basis_A[n][d][r]
  A1[(long)d * (NBb * Rr) + n * Rr + r] = v; // [D, NB*R]
  A2[(long)(n * Rr + r) * Dd + d]       = v; // [NB*R, D]
}

// ---------------------------------------------------------------------------
// Host side
// ---------------------------------------------------------------------------
static inline void gemm(hipStream_t st,
                        const float* A, long lda, long sA1, long sA2,
                        const float* B, long ldb, long sB1, long sB2, int transB,
                        const float* bias, const float* addend,
                        float* C, long ldc, long sC1, long sC2,
                        int M, int N, int Kd, int Z, int HH, int mode,
                        float scale, const float* scale_ptr) {
  GemmP p{A, lda, sA1, sA2, B, ldb, sB1, sB2, bias, addend,
          C, ldc, sC1, sC2, M, N, Kd, HH, transB, mode, scale, scale_ptr};
  dim3 grid(M / 64, N / 64, Z);
  gemm_f32_wmma<<<grid, 256, 0, st>>>(p);
}

extern "C" void kernel_launch(void* const* d_in, const int* in_sizes, int n_in,
                              void* d_out, int out_size, void* d_ws, size_t ws_size,
                              hipStream_t stream) {
  (void)in_sizes; (void)n_in; (void)out_size; (void)ws_size;

  const int*   ids       = (const int*)  d_in[0];
  const float* token_emb = (const float*)d_in[1];
  const float* pos_emb   = (const float*)d_in[2];
  const float* basis_A   = (const float*)d_in[3];
  const float* basis_emb = (const float*)d_in[4];
  const float* recipe    = (const float*)d_in[5];
  const float* Wq    = (const float*)d_in[6];   const float* bq    = (const float*)d_in[7];
  const float* Wk    = (const float*)d_in[8];   const float* bk    = (const float*)d_in[9];
  const float* Wv    = (const float*)d_in[10];  const float* bv    = (const float*)d_in[11];
  const float* Ws    = (const float*)d_in[12];  const float* bsv   = (const float*)d_in[13];
  const float* Wup   = (const float*)d_in[14];  const float* bup   = (const float*)d_in[15];
  const float* Wdown = (const float*)d_in[16];  const float* bdown = (const float*)d_in[17];
  const float* ln1_g = (const float*)d_in[18];  const float* ln1_b = (const float*)d_in[19];
  const float* ln2_g = (const float*)d_in[20];  const float* ln2_b = (const float*)d_in[21];
  const float* alpha = (const float*)d_in[22];
  const float* lnf_g = (const float*)d_in[23];  const float* lnf_b = (const float*)d_in[24];
  float* out = (float*)d_out;

  // ---- workspace carve (~70 MB of f32) ----
  float* ws = (float*)d_ws;
  auto alloc = [&](size_t n) { float* p = ws; ws += (n + 63) & ~(size_t)63; return p; };
  float* x      = alloc((size_t)TOK * Dd);
  float* n1     = alloc((size_t)TOK * Dd);
  float* n2     = alloc((size_t)TOK * Dd);
  float* qb     = alloc((size_t)TOK * Dd);
  float* kb2    = alloc((size_t)TOK * Dd);
  float* vb     = alloc((size_t)TOK * Dd);
  float* ctx    = alloc((size_t)TOK * Dd);
  float* query  = alloc((size_t)TOK * Dd);
  float* xf     = alloc((size_t)TOK * Dd);
  float* attn   = alloc((size_t)Bb * Hh * Ss * Ss);
  float* scores = alloc((size_t)TOK * Nn);
  float* wts    = alloc((size_t)TOK * Kk);
  int*   tidx   = (int*)alloc((size_t)TOK * Kk);
  float* softrec= alloc((size_t)Nn * NBb);
  float* ne     = alloc((size_t)Nn * Dd);
  float* tr     = alloc((size_t)TOK * NBb);
  float* proj   = alloc((size_t)TOK * NBb * Rr);
  float* hbuf   = alloc((size_t)TOK * Rr);
  float* gbuf   = alloc((size_t)TOK * NBb * Rr);
  float* ff1    = alloc((size_t)TOK * DFFd);
  float* A1     = alloc((size_t)Dd * NBb * Rr);
  float* A2     = alloc((size_t)Dd * NBb * Rr);

  const long SD = (long)Ss * Dd, SS = (long)Ss * Ss;

  // ---- prologue ----
  basisA_transpose_kernel<<<(NBb * Dd * Rr) / 256, 256, 0, stream>>>(basis_A, A1, A2);
  embed_kernel<<<(TOK * Dd) / 256, 256, 0, stream>>>(ids, token_emb, pos_emb, x);

  for (int i = 0; i < Ll; ++i) {
    const float* Wq_i = Wq + (size_t)i * Dd * Dd;      const float* bq_i = bq + (size_t)i * Dd;
    const float* Wk_i = Wk + (size_t)i * Dd * Dd;      const float* bk_i = bk + (size_t)i * Dd;
    const float* Wv_i = Wv + (size_t)i * Dd * Dd;      const float* bv_i = bv + (size_t)i * Dd;
    const float* Ws_i = Ws + (size_t)i * 2 * Dd * Dd;  const float* bs_i = bsv + (size_t)i * Dd;
    const float* Wu_i = Wup + (size_t)i * Dd * DFFd;   const float* bu_i = bup + (size_t)i * DFFd;
    const float* Wd_i = Wdown + (size_t)i * DFFd * Dd; const float* bd_i = bdown + (size_t)i * Dd;

    // n1 = LN(x)
    ln_kernel<<<TOK, 256, 0, stream>>>(x, ln1_g + (size_t)i * Dd, ln1_b + (size_t)i * Dd, n1);
    // q/k/v = n1 @ W + b
    gemm(stream, n1, Dd,0,0, Wq_i, Dd,0,0,0, bq_i, nullptr, qb,  Dd,0,0, TOK, Dd, Dd, 1,1, 0, 1.f, nullptr);
    gemm(stream, n1, Dd,0,0, Wk_i, Dd,0,0,0, bk_i, nullptr, kb2, Dd,0,0, TOK, Dd, Dd, 1,1, 0, 1.f, nullptr);
    gemm(stream, n1, Dd,0,0, Wv_i, Dd,0,0,0, bv_i, nullptr, vb,  Dd,0,0, TOK, Dd, Dd, 1,1, 0, 1.f, nullptr);
    // attn = mask(q @ k^T * 1/sqrt(dh)), batched over (b,h)
    gemm(stream, qb, Dd, SD, DHd, kb2, Dd, SD, DHd, /*transB=*/1, nullptr, nullptr,
         attn, Ss, (long)Hh * SS, SS, Ss, Ss, DHd, Bb * Hh, Hh, /*mode=*/2, 0.125f, nullptr);
    softmax512_kernel<<<Bb * Hh * Ss, 256, 0, stream>>>(attn);
    // ctx = attn @ v
    gemm(stream, attn, Ss, (long)Hh * SS, SS, vb, Dd, SD, DHd, 0, nullptr, nullptr,
         ctx, Dd, SD, DHd, Ss, DHd, Ss, Bb * Hh, Hh, 0, 1.f, nullptr);
    // query = n1 @ Ws[:D] + bs ; query += ctx @ Ws[D:]
    gemm(stream, n1,  Dd,0,0, Ws_i,               Dd,0,0,0, bs_i,   nullptr, query, Dd,0,0, TOK, Dd, Dd, 1,1, 0, 1.f, nullptr);
    gemm(stream, ctx, Dd,0,0, Ws_i + (size_t)Dd*Dd, Dd,0,0,0, nullptr, query,  query, Dd,0,0, TOK, Dd, Dd, 1,1, 0, 1.f, nullptr);
    // routing embeddings + scores + top-k
    softrec_kernel<<<Nn, 32, 0, stream>>>(recipe + (size_t)i * Nn * NBb, softrec);
    ne_kernel<<<(Nn * Dd) / 256, 256, 0, stream>>>(softrec, basis_emb, ne);
    gemm(stream, query, Dd,0,0, ne, Dd,0,0, /*transB=*/1, nullptr, nullptr,
         scores, Nn,0,0, TOK, Nn, Dd, 1,1, 0, 1.f, nullptr);
    topk_kernel<<<TOK / 256, 256, 0, stream>>>(scores, wts, tidx);
    token_recipe_kernel<<<(TOK * NBb) / 256, 256, 0, stream>>>(wts, tidx, softrec, tr);
    // n2 = LN(x); basis residual delta
    ln_kernel<<<TOK, 256, 0, stream>>>(x, ln2_g + (size_t)i * Dd, ln2_b + (size_t)i * Dd, n2);
    gemm(stream, n2, Dd,0,0, A1, NBb * Rr,0,0,0, nullptr, nullptr,
         proj, NBb * Rr,0,0, TOK, NBb * Rr, Dd, 1,1, 0, 1.f, nullptr);
    h_kernel<<<(TOK * Rr) / 256, 256, 0, stream>>>(tr, proj, hbuf);
    g_kernel<<<(TOK * NBb * Rr) / 256, 256, 0, stream>>>(tr, hbuf, gbuf);
    // xf = n2 + alpha[i] * (g @ A2)     (alpha read on-device via scale_ptr)
    gemm(stream, gbuf, NBb * Rr,0,0, A2, Dd,0,0,0, nullptr, n2,
         xf, Dd,0,0, TOK, Dd, NBb * Rr, 1,1, 0, 1.f, alpha + i);
    // ff = gelu(xf @ Wup + bup) @ Wdown + bdown ; x += ff
    gemm(stream, xf,  Dd,0,0,   Wu_i, DFFd,0,0,0, bu_i, nullptr, ff1, DFFd,0,0, TOK, DFFd, Dd,  1,1, /*gelu*/1, 1.f, nullptr);
    gemm(stream, ff1, DFFd,0,0, Wd_i, Dd,0,0,0,   bd_i, x,       x,   Dd,0,0,   TOK, Dd, DFFd, 1,1, 0, 1.f, nullptr);
  }

  // ---- final LN + tied head ----
  ln_kernel<<<TOK, 256, 0, stream>>>(x, lnf_g, lnf_b, xf);
  gemm(stream, xf, Dd,0,0, token_emb, Dd,0,0, /*transB=*/1, nullptr, nullptr,
       out, Vv,0,0, TOK, Vv, Dd, 1,1, 0, 1.f, nullptr);
}